// Decoder_35734127903141
// MI455X (gfx1250) — compile-verified
//
#include <hip/hip_runtime.h>
#include <hip/hip_bf16.h>

// ---------------------------------------------------------------------------
// Show-attend-tell decoder for MI455X (gfx1250, wave32, WMMA).
// All GEMMs use V_WMMA_F32_16X16X4_F32 (native fp32 matrix path, fp32 accum).
// The 512x32000 vocab projection is hoisted out of the 39-step loop into one
// 1248x32000x512 GEMM over the stored hidden states (W_fc stays L2-resident).
// ---------------------------------------------------------------------------

#define T_STEPS 39
#define BATCH   32
#define VOCAB   32000
#define PIX     196
#define ENCD    2048
#define AD      512
#define DD      512
#define ED      512
#define CAPL    40

typedef __attribute__((ext_vector_type(2))) float v2f;
typedef __attribute__((ext_vector_type(8))) float v8f;

#define BM    64
#define BN    128
#define KTILE 16

__device__ __forceinline__ float sigmoidf_(float x) {
    return 1.0f / (1.0f + __expf(-x));
}

// ----------------------------------------------------------------------------
// WMMA fp32 GEMM core: one 256-thread block computes a BM x BN tile of
// C = A(MxK) @ B(KxN). 8 waves arranged 4(M) x 2(N); each wave owns a 16x64
// strip (4 v8f accumulators), reusing its A fragment across 4 WMMAs.
// K must be a multiple of KTILE; N a multiple of BN; M is guarded.
// ----------------------------------------------------------------------------
__device__ __forceinline__ void gemm_core_f32(
    const float* __restrict__ A, const float* __restrict__ B,
    int M, int N, int K, int m0, int n0, v8f acc[4])
{
    __shared__ float As[KTILE][BM];   // [k][m]  (transposed for conflict-free A reads)
    __shared__ float Bs[KTILE][BN];   // [k][n]

    const int tid  = threadIdx.x;
    const int lane = tid & 31;
    const int wave = tid >> 5;
    const int hl   = lane >> 4;       // half of wave: selects K pair
    const int ln   = lane & 15;
    const int wm   = wave & 3;        // 4 M-subtiles of 16
    const int wn   = wave >> 2;       // 2 N-subtiles of 64

    for (int k0 = 0; k0 < K; k0 += KTILE) {
        {   // stage A tile (64 x 16), zero-padded past M
            const int j  = tid & 15;
            const int ib = tid >> 4;
#pragma unroll
            for (int it = 0; it < 4; ++it) {
                const int i  = ib + it * 16;
                const int gm = m0 + i;
                As[j][i] = (gm < M) ? A[(size_t)gm * K + (k0 + j)] : 0.0f;
            }
        }
        {   // stage B tile (16 x 128)
            const int nn = tid & 127;
            const int jb = tid >> 7;
#pragma unroll
            for (int it = 0; it < 8; ++it) {
                const int j = jb + it * 2;
                Bs[j][nn] = B[(size_t)(k0 + j) * N + (n0 + nn)];
            }
        }
        __syncthreads();
#pragma unroll
        for (int kk = 0; kk < KTILE; kk += 4) {
            // A 16x4 fragment: lanes 0-15 hold (K0,K1), lanes 16-31 hold (K2,K3)
            v2f av;
            av.x = As[kk + 2 * hl + 0][wm * 16 + ln];
            av.y = As[kk + 2 * hl + 1][wm * 16 + ln];
#pragma unroll
            for (int s = 0; s < 4; ++s) {
                v2f bv;     // B 4x16 fragment, symmetric layout
                bv.x = Bs[kk + 2 * hl + 0][wn * 64 + s * 16 + ln];
                bv.y = Bs[kk + 2 * hl + 1][wn * 64 + s * 16 + ln];
                acc[s] = __builtin_amdgcn_wmma_f32_16x16x4_f32(
                    false, av, false, bv, (short)0, acc[s], false, false);
            }
        }
        __syncthreads();
    }
}

// EPI: 0 = C = acc + bias ; 1 = sigmoid(acc + bias) ; 2 = C += acc + bias
template <int EPI>
__global__ __launch_bounds__(256) void Decoder_gemm_kernel(
    const float* __restrict__ A, const float* __restrict__ B,
    const float* __restrict__ bias, float* __restrict__ C,
    int M, int N, int K)
{
    v8f acc[4];
#pragma unroll
    for (int s = 0; s < 4; ++s) acc[s] = (v8f){0,0,0,0,0,0,0,0};

    const int m0 = blockIdx.y * BM;
    const int n0 = blockIdx.x * BN;
    gemm_core_f32(A, B, M, N, K, m0, n0, acc);

    const int tid = threadIdx.x, lane = tid & 31, wave = tid >> 5;
    const int hl = lane >> 4, ln = lane & 15, wm = wave & 3, wn = wave >> 2;
#pragma unroll
    for (int s = 0; s < 4; ++s) {
        const int col = n0 + wn * 64 + s * 16 + ln;
        const float bv = bias ? bias[col] : 0.0f;
#pragma unroll
        for (int r = 0; r < 8; ++r) {
            const int row = m0 + wm * 16 + r + 8 * hl;   // C layout: M = r + 8*half
            if (row < M) {
                const size_t idx = (size_t)row * N + col;
                float v = acc[s][r] + bv;
                if (EPI == 1) v = sigmoidf_(v);
                else if (EPI == 2) v += C[idx];
                C[idx] = v;
            }
        }
    }
}

// Final vocab GEMM: A = Hstore (row r = t*B + b), out[b][t][v] = (acc+b_fc[v])*mask[r]
__global__ __launch_bounds__(256) void Decoder_vocab_kernel(
    const float* __restrict__ Hst, const float* __restrict__ Wfc,
    const float* __restrict__ bfc, const float* __restrict__ maskTB,
    float* __restrict__ preds, int M, int N, int K)
{
    v8f acc[4];
#pragma unroll
    for (int s = 0; s < 4; ++s) acc[s] = (v8f){0,0,0,0,0,0,0,0};

    const int m0 = blockIdx.y * BM;
    const int n0 = blockIdx.x * BN;
    gemm_core_f32(Hst, Wfc, M, N, K, m0, n0, acc);

    const int tid = threadIdx.x, lane = tid & 31, wave = tid >> 5;
    const int hl = lane >> 4, ln = lane & 15, wm = wave & 3, wn = wave >> 2;
#pragma unroll
    for (int s = 0; s < 4; ++s) {
        const int col = n0 + wn * 64 + s * 16 + ln;
        const float bv = bfc[col];
#pragma unroll
        for (int r = 0; r < 8; ++r) {
            const int row = m0 + wm * 16 + r + 8 * hl;
            if (row < M) {
                const int t = row >> 5;          // B == 32
                const int b = row & 31;
                const float v = (acc[s][r] + bv) * maskTB[row];
                preds[((size_t)b * T_STEPS + t) * VOCAB + col] = v;
            }
        }
    }
}

// avg[b][e] = mean_p encoder_out[b][p][e]
__global__ __launch_bounds__(256) void Decoder_avg_kernel(
    const float* __restrict__ enc, float* __restrict__ avg)
{
    const int b = blockIdx.x;
    for (int e = threadIdx.x; e < ENCD; e += 256) {
        const float* p = enc + (size_t)b * PIX * ENCD + e;
        float s = 0.0f;
        for (int pp = 0; pp < PIX; ++pp) s += p[(size_t)pp * ENCD];
        avg[b * ENCD + e] = s * (1.0f / (float)PIX);
    }
}

// Per-step attention + softmax + context + x-assembly. One block per batch row.
__global__ __launch_bounds__(256) void Decoder_attn_kernel(
    const float* __restrict__ enc_att, const float* __restrict__ dec,
    const float* __restrict__ w_att, const float* __restrict__ b_att,
    const float* __restrict__ encoder_out, const float* __restrict__ gate,
    const float* __restrict__ embedding, const int* __restrict__ captions,
    const int* __restrict__ caplens,
    float* __restrict__ xbuf, float* __restrict__ alphas_out,
    float* __restrict__ maskTB, int t)
{
    const int b   = blockIdx.x;
    const int tid = threadIdx.x;
    const int lane = tid & 31, wave = tid >> 5;

    __shared__ float s_dec[AD];
    __shared__ float s_w[AD];
    __shared__ float s_e[224];
    __shared__ float s_alpha[224];
    __shared__ float s_red[8];

    for (int i = tid; i < AD; i += 256) {
        s_dec[i] = dec[b * AD + i];
        s_w[i]   = w_att[i];
    }
    __syncthreads();

    // e[p] = relu(enc_att[b][p][:] + dec[b][:]) . w_att   (one wave per pixel)
    for (int p = wave; p < PIX; p += 8) {
        const float* ea = enc_att + ((size_t)b * PIX + p) * AD;
        float sum = 0.0f;
        for (int a = lane; a < AD; a += 32) {
            float v = ea[a] + s_dec[a];
            v = v > 0.0f ? v : 0.0f;
            sum += v * s_w[a];
        }
        for (int off = 16; off > 0; off >>= 1) sum += __shfl_xor(sum, off, 32);
        if (lane == 0) s_e[p] = sum + b_att[0];
    }
    __syncthreads();

    // softmax over 196
    const float v0 = (tid < PIX) ? s_e[tid] : -1e30f;
    float mx = v0;
    for (int off = 16; off > 0; off >>= 1) mx = fmaxf(mx, __shfl_xor(mx, off, 32));
    if (lane == 0) s_red[wave] = mx;
    __syncthreads();
    if (tid == 0) {
        float m2 = s_red[0];
        for (int i = 1; i < 8; ++i) m2 = fmaxf(m2, s_red[i]);
        s_red[0] = m2;
    }
    __syncthreads();
    const float gmax = s_red[0];
    const float ex = (tid < PIX) ? __expf(v0 - gmax) : 0.0f;
    float sm = ex;
    for (int off = 16; off > 0; off >>= 1) sm += __shfl_xor(sm, off, 32);
    __syncthreads();                       // everyone has read s_red[0]
    if (lane == 0) s_red[wave] = sm;
    __syncthreads();
    if (tid == 0) {
        float s2 = 0.0f;
        for (int i = 0; i < 8; ++i) s2 += s_red[i];
        s_red[0] = s2;
    }
    __syncthreads();
    const float inv = 1.0f / s_red[0];

    const int active = (caplens[b] - 1) > t;
    if (tid < PIX) {
        const float al = ex * inv;
        s_alpha[tid] = al;
        alphas_out[((size_t)b * T_STEPS + t) * PIX + tid] = active ? al : 0.0f;
    }
    if (tid == 0) maskTB[t * BATCH + b] = active ? 1.0f : 0.0f;
    __syncthreads();

    // x[b][0:512] = embedding[caption[b][t]]
    const int cap = captions[b * CAPL + t];
    for (int i = tid; i < ED; i += 256)
        xbuf[(size_t)b * (ED + ENCD) + i] = embedding[(size_t)cap * ED + i];

    // x[b][512:2560] = gate * (alpha . encoder_out)
    for (int e = tid; e < ENCD; e += 256) {
        const float* eo = encoder_out + (size_t)b * PIX * ENCD + e;
        float sum = 0.0f;
#pragma unroll 4
        for (int p = 0; p < PIX; ++p) sum += s_alpha[p] * eo[(size_t)p * ENCD];
        xbuf[(size_t)b * (ED + ENCD) + ED + e] = gate[(size_t)b * ENCD + e] * sum;
    }
}

// LSTM gating + masked state update + hidden-state history
__global__ __launch_bounds__(256) void Decoder_lstm_kernel(
    const float* __restrict__ g, float* __restrict__ h, float* __restrict__ c,
    float* __restrict__ Hst, const int* __restrict__ caplens, int t)
{
    const int idx = blockIdx.x * 256 + threadIdx.x;   // B*D = 16384
    const int b = idx >> 9, d = idx & 511;
    const float* gb = g + (size_t)b * (4 * DD);
    const float iv = sigmoidf_(gb[d]);
    const float fv = sigmoidf_(gb[DD + d]);
    const float gv = tanhf(gb[2 * DD + d]);
    const float ov = sigmoidf_(gb[3 * DD + d]);
    const float cn = fv * c[idx] + iv * gv;
    const float hn = ov * tanhf(cn);
    Hst[((size_t)t * BATCH + b) * DD + d] = hn;       // preds mask applied in epilogue
    if ((caplens[b] - 1) > t) { h[idx] = hn; c[idx] = cn; }
}

extern "C" void kernel_launch(void* const* d_in, const int* in_sizes, int n_in,
                              void* d_out, int out_size, void* d_ws, size_t ws_size,
                              hipStream_t stream) {
    const float* encoder_out = (const float*)d_in[0];
    const int*   captions    = (const int*)d_in[1];
    const int*   caplens     = (const int*)d_in[2];
    const float* embedding   = (const float*)d_in[3];
    const float* W_enc = (const float*)d_in[4];  const float* b_enc = (const float*)d_in[5];
    const float* W_dec = (const float*)d_in[6];  const float* b_dec = (const float*)d_in[7];
    const float* w_att = (const float*)d_in[8];  const float* b_att = (const float*)d_in[9];
    const float* W_ih  = (const float*)d_in[10]; const float* b_ih  = (const float*)d_in[11];
    const float* W_hh  = (const float*)d_in[12]; const float* b_hh  = (const float*)d_in[13];
    const float* W_h   = (const float*)d_in[14]; const float* b_h   = (const float*)d_in[15];
    const float* W_c   = (const float*)d_in[16]; const float* b_c   = (const float*)d_in[17];
    const float* W_fb  = (const float*)d_in[18]; const float* b_fb  = (const float*)d_in[19];
    const float* W_fc  = (const float*)d_in[20]; const float* b_fc  = (const float*)d_in[21];

    float* preds  = (float*)d_out;                                 // [B,T,V]
    float* alphas = preds + (size_t)BATCH * T_STEPS * VOCAB;       // [B,T,P]

    // workspace carve-out (~16.7 MB)
    float* wsf     = (float*)d_ws;
    float* enc_att = wsf;                       // 6272*512
    float* avg     = enc_att + (size_t)BATCH * PIX * AD;   // 32*2048
    float* h       = avg + (size_t)BATCH * ENCD;           // 32*512
    float* c       = h + (size_t)BATCH * DD;
    float* dec     = c + (size_t)BATCH * DD;
    float* gate    = dec + (size_t)BATCH * AD;             // 32*2048
    float* xbuf    = gate + (size_t)BATCH * ENCD;          // 32*2560
    float* gbuf    = xbuf + (size_t)BATCH * (ED + ENCD);   // 32*2048
    float* Hst     = gbuf + (size_t)BATCH * 4 * DD;        // 1248*512
    float* maskTB  = Hst + (size_t)T_STEPS * BATCH * DD;   // 1248

    // ---- loop-invariant precompute ----
    Decoder_avg_kernel<<<BATCH, 256, 0, stream>>>(encoder_out, avg);
    Decoder_gemm_kernel<0><<<dim3(AD / BN, 1), 256, 0, stream>>>(avg, W_h, b_h, h, BATCH, DD, ENCD);
    Decoder_gemm_kernel<0><<<dim3(AD / BN, 1), 256, 0, stream>>>(avg, W_c, b_c, c, BATCH, DD, ENCD);
    Decoder_gemm_kernel<0><<<dim3(AD / BN, (BATCH * PIX) / BM), 256, 0, stream>>>(
        encoder_out, W_enc, b_enc, enc_att, BATCH * PIX, AD, ENCD);

    // ---- sequential decode loop (critical path; vocab GEMM hoisted out) ----
    for (int t = 0; t < T_STEPS; ++t) {
        Decoder_gemm_kernel<0><<<dim3(AD / BN, 1), 256, 0, stream>>>(
            h, W_dec, b_dec, dec, BATCH, AD, DD);
        Decoder_gemm_kernel<1><<<dim3(ENCD / BN, 1), 256, 0, stream>>>(
            h, W_fb, b_fb, gate, BATCH, ENCD, DD);
        Decoder_attn_kernel<<<BATCH, 256, 0, stream>>>(
            enc_att, dec, w_att, b_att, encoder_out, gate,
            embedding, captions, caplens, xbuf, alphas, maskTB, t);
        Decoder_gemm_kernel<0><<<dim3((4 * DD) / BN, 1), 256, 0, stream>>>(
            xbuf, W_ih, b_ih, gbuf, BATCH, 4 * DD, ED + ENCD);
        Decoder_gemm_kernel<2><<<dim3((4 * DD) / BN, 1), 256, 0, stream>>>(
            h, W_hh, b_hh, gbuf, BATCH, 4 * DD, DD);
        Decoder_lstm_kernel<<<(BATCH * DD) / 256, 256, 0, stream>>>(
            gbuf, h, c, Hst, caplens, t);
    }

    // ---- one fat vocab GEMM: (T*B=1248) x 32000, K=512; W_fc is L2-resident ----
    Decoder_vocab_kernel<<<dim3(VOCAB / BN, (T_STEPS * BATCH + BM - 1) / BM), 256, 0, stream>>>(
        Hst, W_fc, b_fc, maskTB, preds, T_STEPS * BATCH, VOCAB, DD);
}